// ReservoirNet_49194555408948
// MI455X (gfx1250) — compile-verified
//
#include <hip/hip_runtime.h>
#include <math.h>

// ---------------- problem constants ----------------
#define IN_SIZE 64
#define RES     2048
#define OUTSZ   16
#define TS      8192
#define DD      2113          // 1 + IN_SIZE + RES
#define NT      133           // ceil(DD/16)
#define DP      (NT*16)       // 2128, padded D
#define NTRI    (NT*(NT+1)/2) // 8911 lower-triangle tiles
#define ALEAK   0.3f
#define RIDGE   1e-12f

#define SCAN_BLOCKS 256       // 256 blk * 8 waves = 2048 waves = one wave per reservoir row
#define CHOL_BLOCKS 256

typedef float v2f __attribute__((ext_vector_type(2)));
typedef float v8f __attribute__((ext_vector_type(8)));

// ---------------- software grid barrier (persistent kernels) ----------------
__device__ __forceinline__ void grid_sync(unsigned* cnt, volatile unsigned* gen, unsigned nblk) {
    __syncthreads();
    if (threadIdx.x == 0) {
        __threadfence();
        unsigned g = *gen;
        if (atomicAdd(cnt, 1u) == nblk - 1u) {
            *cnt = 0u;
            __threadfence();
            atomicAdd((unsigned*)gen, 1u);
        } else {
            while (*gen == g) { __builtin_amdgcn_s_sleep(2); }
        }
        __threadfence();
    }
    __syncthreads();
}

// ---------------- build X rows: ones, data^T, zero padding ----------------
// X is DP x TS row-major. Row 0 = ones, rows 1..64 = data^T, rows 2113..2127 = 0.
__global__ void fill_x(const float* __restrict__ data, float* __restrict__ Xp) {
    int idx = blockIdx.x * blockDim.x + threadIdx.x;
    if (idx >= 80 * TS) return;
    int rp = idx >> 13;          // 0..79
    int t  = idx & (TS - 1);
    if (rp == 0)        Xp[t] = 1.0f;
    else if (rp <= 64)  Xp[(size_t)rp * TS + t] = data[(size_t)t * IN_SIZE + (rp - 1)];
    else                Xp[(size_t)(DD + (rp - 65)) * TS + t] = 0.0f;
}

// ---------------- reservoir scan: W register-resident, 1 wave per row ----------------
__global__ __launch_bounds__(256) void esn_scan(const float* __restrict__ data,
                                                const float* __restrict__ Win,
                                                const float* __restrict__ W,
                                                float* __restrict__ xbuf,   // 2*RES double buffer
                                                float* __restrict__ Xp,
                                                unsigned* cnt, unsigned* gen) {
    const int lane = threadIdx.x & 31;
    const int row  = (blockIdx.x * blockDim.x + threadIdx.x) >> 5;  // 0..2047 exactly

    // Register-resident row of W: lane holds cols {lane, lane+32, ...} (64 values)
    const float* wrow = W + (size_t)row * RES;
    float wreg[64];
#pragma unroll
    for (int c = 0; c < 64; ++c) wreg[c] = wrow[c * 32 + lane];

    const float bias = Win[(size_t)row * 65];
    const float wi0  = Win[(size_t)row * 65 + 1 + lane];
    const float wi1  = Win[(size_t)row * 65 + 33 + lane];

    float xr = 0.0f;
    for (int t = 0; t < TS; ++t) {
        const float* xin = xbuf + (t & 1) * RES;     // holds x_{t-1}
        const float* u   = data + (size_t)t * IN_SIZE;
        float acc = wi0 * u[lane] + wi1 * u[32 + lane];
#pragma unroll
        for (int c = 0; c < 64; ++c) acc = fmaf(wreg[c], xin[c * 32 + lane], acc);
#pragma unroll
        for (int off = 16; off > 0; off >>= 1) acc += __shfl_xor(acc, off, 32);
        float pre = acc + bias;
        float s   = 1.0f / (1.0f + __expf(-pre));
        xr = (1.0f - ALEAK) * xr + ALEAK * s;
        if (lane == 0) {
            xbuf[((t + 1) & 1) * RES + row] = xr;            // publish x_t
            Xp[(size_t)(65 + row) * TS + t] = xr;            // xs^T row
        }
        grid_sync(cnt, gen, SCAN_BLOCKS);                    // x_t visible before step t+1
    }
}

// ---------------- Gram: G = X X^T lower-triangle tiles via WMMA f32 16x16x4 ----------------
__global__ __launch_bounds__(128) void gram_wmma(const float* __restrict__ Xp,
                                                 float* __restrict__ G) {
    const int gw   = (blockIdx.x * blockDim.x + threadIdx.x) >> 5;
    if (gw >= NTRI) return;                                  // wave-uniform exit (EXEC stays full)
    const int lane = threadIdx.x & 31;
    const int ll   = lane & 15, lh = lane >> 4;

    // linear lower-triangle index -> (r, c), c <= r
    int r = (int)((sqrtf(8.0f * gw + 1.0f) - 1.0f) * 0.5f);
    while ((r + 1) * (r + 2) / 2 <= gw) ++r;
    while (r * (r + 1) / 2 > gw) --r;
    const int c  = gw - r * (r + 1) / 2;
    const int i0 = r * 16, j0 = c * 16;

    // A(16x4): lane(M=ll) vgpr pair = X[i0+ll][k+2lh .. +1]; B(4x16): X[j0+ll][k+2lh .. +1]
    const float* arow = Xp + (size_t)(i0 + ll) * TS;
    const float* brow = Xp + (size_t)(j0 + ll) * TS;

    v8f acc = {};
    for (int k = 0; k < TS; k += 16) {
        __builtin_prefetch(arow + k + 256, 0, 0);            // global_prefetch_b8 ahead of WMMA
        __builtin_prefetch(brow + k + 256, 0, 0);
#pragma unroll
        for (int kk = 0; kk < 16; kk += 4) {
            v2f a = *(const v2f*)(arow + k + kk + 2 * lh);
            v2f b = *(const v2f*)(brow + k + kk + 2 * lh);
            acc = __builtin_amdgcn_wmma_f32_16x16x4_f32(false, a, false, b,
                                                        (short)0, acc, false, false);
        }
    }
    // C/D layout: c[vr] -> G[i0 + vr + 8*lh][j0 + ll]
    float* gp = G + (size_t)(i0 + 8 * lh) * DP + j0 + ll;
#pragma unroll
    for (int vr = 0; vr < 8; ++vr) gp[(size_t)vr * DP] = acc[vr];
}

// ---------------- ridge on diagonal ----------------
__global__ void fix_diag(float* __restrict__ G) {
    for (int d = threadIdx.x; d < DD; d += blockDim.x)
        G[(size_t)d * DP + d] += RIDGE;
}

// ---------------- Z = X * labels  (D x 16), i.e. (labels^T X^T)^T ----------------
__global__ void rhs_build(const float* __restrict__ labels, const float* __restrict__ Xp,
                          float* __restrict__ Z) {
    int idx = blockIdx.x * blockDim.x + threadIdx.x;
    if (idx >= DD * OUTSZ) return;
    int d = idx >> 4, o = idx & 15;
    const float* xr = Xp + (size_t)d * TS;
    float s = 0.0f;
    for (int t = 0; t < TS; ++t) s = fmaf(xr[t], labels[(size_t)t * OUTSZ + o], s);
    Z[(size_t)d * OUTSZ + o] = s;
}

// ---------------- blocked Cholesky (NB=16), trailing update via WMMA ----------------
__global__ __launch_bounds__(256) void chol_factor(float* __restrict__ G,
                                                   unsigned* cnt, unsigned* gen) {
    const int tid    = threadIdx.x;
    const int gtid   = blockIdx.x * blockDim.x + tid;
    const int lane   = tid & 31;
    const int gw     = gtid >> 5;
    const int nwaves = (CHOL_BLOCKS * 256) >> 5;
    const int ll     = lane & 15, lh = lane >> 4;

    for (int kb = 0; kb < NT; ++kb) {
        const int k0 = kb * 16;

        // phase 1: 16x16 diagonal block Cholesky (serial, tiny)
        if (blockIdx.x == 0 && tid == 0) {
            for (int j = 0; j < 16; ++j) {
                int cj = k0 + j; if (cj >= DD) break;
                float s = G[(size_t)cj * DP + cj];
                for (int m = 0; m < j; ++m) { float v = G[(size_t)cj * DP + k0 + m]; s -= v * v; }
                s = sqrtf(fmaxf(s, 1e-30f));
                G[(size_t)cj * DP + cj] = s;
                float inv = 1.0f / s;
                for (int i = j + 1; i < 16; ++i) {
                    int ci = k0 + i; if (ci >= DD) break;
                    float t2 = G[(size_t)ci * DP + cj];
                    for (int m = 0; m < j; ++m)
                        t2 -= G[(size_t)ci * DP + k0 + m] * G[(size_t)cj * DP + k0 + m];
                    G[(size_t)ci * DP + cj] = t2 * inv;
                }
            }
        }
        grid_sync(cnt, gen, CHOL_BLOCKS);

        // phase 2: panel solve, one thread per trailing row
        for (int i = k0 + 16 + gtid; i < DD; i += CHOL_BLOCKS * 256) {
            float* gi = G + (size_t)i * DP + k0;
            const float* gd = G + (size_t)k0 * DP + k0;
#pragma unroll
            for (int j = 0; j < 16; ++j) {
                if (k0 + j >= DD) break;
                float s = gi[j];
                for (int m = 0; m < j; ++m) s -= gi[m] * gd[(size_t)j * DP + m];
                gi[j] = s / gd[(size_t)j * DP + j];
            }
        }
        grid_sync(cnt, gen, CHOL_BLOCKS);

        // phase 3: rank-16 trailing update on lower tiles, C -= A * B^T via WMMA
        const int m  = NT - 1 - kb;
        const int nt = m * (m + 1) / 2;
        for (int tIdx = gw; tIdx < nt; tIdx += nwaves) {
            int r = (int)((sqrtf(8.0f * tIdx + 1.0f) - 1.0f) * 0.5f);
            while ((r + 1) * (r + 2) / 2 <= tIdx) ++r;
            while (r * (r + 1) / 2 > tIdx) --r;
            int cc2 = tIdx - r * (r + 1) / 2;
            int i0 = (kb + 1 + r) * 16, j0 = (kb + 1 + cc2) * 16;
            const float* ap = G + (size_t)(i0 + ll) * DP + k0;
            const float* bp = G + (size_t)(j0 + ll) * DP + k0;
            float* gp = G + (size_t)(i0 + 8 * lh) * DP + j0 + ll;
            v8f acc;
#pragma unroll
            for (int vr = 0; vr < 8; ++vr) acc[vr] = gp[(size_t)vr * DP];
#pragma unroll
            for (int kk = 0; kk < 16; kk += 4) {
                v2f a = *(const v2f*)(ap + kk + 2 * lh);
                v2f b = *(const v2f*)(bp + kk + 2 * lh);
                a = -a;   // f32 WMMA NEG bits are C-only (ISA 7.12): negate A in VALU
                acc = __builtin_amdgcn_wmma_f32_16x16x4_f32(false, a, false, b,
                                                            (short)0, acc, false, false);
            }
#pragma unroll
            for (int vr = 0; vr < 8; ++vr) gp[(size_t)vr * DP] = acc[vr];
        }
        grid_sync(cnt, gen, CHOL_BLOCKS);
    }
}

// ---------------- triangular solves: L y = Z, L^T z = y (16 RHS, one block) ----------------
__global__ __launch_bounds__(256) void tri_solve(const float* __restrict__ G,
                                                 float* __restrict__ Z) {
    const int tid = threadIdx.x;
    const int o = tid & 15, rr = tid >> 4;        // 16 rows x 16 RHS per pass
    for (int j = 0; j < DD; ++j) {                // forward
        if (tid < 16) Z[(size_t)j * 16 + tid] /= G[(size_t)j * DP + j];
        __syncthreads();
        float zj = Z[(size_t)j * 16 + o];
        for (int i = j + 1 + rr; i < DD; i += 16)
            Z[(size_t)i * 16 + o] -= G[(size_t)i * DP + j] * zj;
        __syncthreads();
    }
    for (int j = DD - 1; j >= 0; --j) {           // backward (L^T)
        if (tid < 16) Z[(size_t)j * 16 + tid] /= G[(size_t)j * DP + j];
        __syncthreads();
        float zj = Z[(size_t)j * 16 + o];
        for (int i = rr; i < j; i += 16)
            Z[(size_t)i * 16 + o] -= G[(size_t)j * DP + i] * zj;
        __syncthreads();
    }
}

// ---------------- Wout[o][d] = Z[d][o] ----------------
__global__ void write_out(const float* __restrict__ Z, float* __restrict__ out) {
    int idx = blockIdx.x * blockDim.x + threadIdx.x;
    if (idx >= OUTSZ * DD) return;
    int o = idx / DD, d = idx - o * DD;
    out[idx] = Z[(size_t)d * OUTSZ + o];
}

// ---------------- host launch ----------------
extern "C" void kernel_launch(void* const* d_in, const int* in_sizes, int n_in,
                              void* d_out, int out_size, void* d_ws, size_t ws_size,
                              hipStream_t stream) {
    const float* data   = (const float*)d_in[0];   // [T, 64]
    const float* labels = (const float*)d_in[1];   // [T, 16]
    const float* Win    = (const float*)d_in[2];   // [2048, 65]
    const float* W      = (const float*)d_in[3];   // [2048, 2048]

    char* ws = (char*)d_ws;
    size_t off = 0;
    auto carve = [&](size_t bytes) { size_t p = off; off += (bytes + 255) & ~(size_t)255; return p; };
    unsigned* barrier = (unsigned*)(ws + carve(256));                         // [cnt, gen]
    float* xbuf = (float*)(ws + carve((size_t)2 * RES * sizeof(float)));      // double buffer
    float* Z    = (float*)(ws + carve((size_t)DP * OUTSZ * sizeof(float)));
    float* G    = (float*)(ws + carve((size_t)DP * DP * sizeof(float)));      // ~18 MB
    float* Xp   = (float*)(ws + carve((size_t)DP * TS * sizeof(float)));      // ~70 MB
    (void)ws_size; (void)in_sizes; (void)n_in; (void)out_size;

    hipMemsetAsync(barrier, 0, 256, stream);
    hipMemsetAsync(xbuf, 0, (size_t)2 * RES * sizeof(float), stream);

    fill_x<<<(80 * TS + 255) / 256, 256, 0, stream>>>(data, Xp);
    esn_scan<<<SCAN_BLOCKS, 256, 0, stream>>>(data, Win, W, xbuf, Xp, barrier, barrier + 1);
    gram_wmma<<<(NTRI + 3) / 4, 128, 0, stream>>>(Xp, G);
    fix_diag<<<1, 256, 0, stream>>>(G);
    rhs_build<<<(DD * OUTSZ + 255) / 256, 256, 0, stream>>>(labels, Xp, Z);
    chol_factor<<<CHOL_BLOCKS, 256, 0, stream>>>(G, barrier, barrier + 1);
    tri_solve<<<1, 256, 0, stream>>>(G, Z);
    write_out<<<(OUTSZ * DD + 255) / 256, 256, 0, stream>>>(Z, (float*)d_out);
}